// SPGATConv_16484084483541
// MI455X (gfx1250) — compile-verified
//
#include <hip/hip_runtime.h>
#include <math.h>

typedef __attribute__((ext_vector_type(2))) float v2f;
typedef __attribute__((ext_vector_type(8))) float v8f;

#define H_HEADS 4
#define F_DIM 32
#define C_DIM 128   // H*F == IN == 128
#define NEG_SLOPE 0.2f

// ---- ordered-uint float encoding for atomic max (monotone) ----
__device__ __forceinline__ unsigned enc_ord(float f) {
    unsigned u = __float_as_uint(f);
    return (u & 0x80000000u) ? ~u : (u | 0x80000000u);
}
__device__ __forceinline__ float dec_ord(unsigned u) {
    return (u & 0x80000000u) ? __uint_as_float(u ^ 0x80000000u)
                             : __uint_as_float(~u);
}
#define ENC_NEG_INF 0x007FFFFFu   // enc(-inf)

// ============================================================
// K1: feat_src = feat @ W^T via V_WMMA_F32_16X16X4_F32.
// Block = 256 threads = 8 waves; each wave computes 16 rows x 128 cols.
// W (128x128 f32, 64KB) staged in LDS; A rows held in registers.
// ============================================================
__global__ __launch_bounds__(256) void gat_gemm(const float* __restrict__ feat,
                                                const float* __restrict__ Wfc,
                                                float* __restrict__ feat_src,
                                                int N) {
    __shared__ float Wl[C_DIM * C_DIM];  // 64 KB
    int t = threadIdx.x;
    for (int i = t; i < C_DIM * C_DIM; i += 256) Wl[i] = Wfc[i];
    __syncthreads();

    int wave = t >> 5;
    int lane = t & 31;
    int rowBase = (blockIdx.x * 8 + wave) * 16;
    if (rowBase >= N) return;           // wave-uniform: EXEC stays all-ones

    int lm = lane & 15;
    int kh = (lane >> 4) * 2;           // lanes 0-15 hold K,K+1 ; lanes 16-31 hold K+2,K+3
    int arow = rowBase + lm;
    if (arow >= N) arow = N - 1;        // clamp loads; stores are guarded below
    const float* aptr = feat + (size_t)arow * C_DIM;

    // Preload this wave's A rows: 32 K-steps x 2 floats per lane
    v2f areg[32];
#pragma unroll
    for (int kk = 0; kk < 32; ++kk)
        areg[kk] = *(const v2f*)(aptr + kk * 4 + kh);

    for (int nt = 0; nt < 8; ++nt) {    // 8 column tiles of 16
        int colBase = nt * 16;
        // B[k][n] = W[n][k]; lane's column n = colBase+lm
        const float* bptr = Wl + (size_t)(colBase + lm) * C_DIM;
        v8f c = {0.f, 0.f, 0.f, 0.f, 0.f, 0.f, 0.f, 0.f};
#pragma unroll
        for (int kk = 0; kk < 32; ++kk) {
            v2f b = *(const v2f*)(bptr + kk * 4 + kh);
            c = __builtin_amdgcn_wmma_f32_16x16x4_f32(
                    false, areg[kk], false, b, (short)0, c, false, false);
        }
        // D layout: VGPR r -> row rowBase + r + 8*(lane>=16), col colBase + (lane&15)
        int rrow = rowBase + (lane >> 4) * 8;
#pragma unroll
        for (int r = 0; r < 8; ++r) {
            int row = rrow + r;
            if (row < N)
                feat_src[(size_t)row * C_DIM + colBase + lm] = c[r];
        }
    }
}

// ============================================================
// K2: el[n,h] = feat_src[n,h,:]·attn_l[h,:]; er for first num_dst nodes.
// ============================================================
__global__ __launch_bounds__(256) void gat_attn(const float* __restrict__ feat_src,
                                                const float* __restrict__ attn_l,
                                                const float* __restrict__ attn_r,
                                                float* __restrict__ el,
                                                float* __restrict__ er,
                                                int N, int num_dst) {
    int i = blockIdx.x * blockDim.x + threadIdx.x;
    if (i >= N * H_HEADS) return;
    int n = i >> 2, h = i & 3;
    const float* fs = feat_src + (size_t)n * C_DIM + h * F_DIM;
    const float* al = attn_l + h * F_DIM;
    float sl = 0.f;
#pragma unroll
    for (int f = 0; f < F_DIM; ++f) sl = fmaf(fs[f], al[f], sl);
    el[i] = sl;
    if (n < num_dst) {
        const float* ar = attn_r + h * F_DIM;
        float sr = 0.f;
#pragma unroll
        for (int f = 0; f < F_DIM; ++f) sr = fmaf(fs[f], ar[f], sr);
        er[i] = sr;
    }
}

// ============================================================
// K3: init out=0, m=enc(-inf), denom=0
// ============================================================
__global__ __launch_bounds__(256) void gat_init(float* __restrict__ out,
                                                unsigned* __restrict__ menc,
                                                float* __restrict__ denom,
                                                int num_dst) {
    int i = blockIdx.x * blockDim.x + threadIdx.x;
    int total = num_dst * C_DIM;
    if (i < total) out[i] = 0.f;
    if (i < num_dst * H_HEADS) { menc[i] = ENC_NEG_INF; denom[i] = 0.f; }
}

// ============================================================
// K4 (edge pass A): segment max of leaky-relu(el[src]+er[dst]) per (dst,h)
// ============================================================
__global__ __launch_bounds__(256) void gat_edge_max(const float* __restrict__ el,
                                                    const float* __restrict__ er,
                                                    const int* __restrict__ src,
                                                    const int* __restrict__ dst,
                                                    unsigned* __restrict__ menc,
                                                    int E) {
    int e = blockIdx.x * blockDim.x + threadIdx.x;
    if (e >= E) return;
    int s = src[e], d = dst[e];
    const float4 l4 = *(const float4*)(el + 4 * (size_t)s);
    const float4 r4 = *(const float4*)(er + 4 * (size_t)d);
    float v[4] = {l4.x + r4.x, l4.y + r4.y, l4.z + r4.z, l4.w + r4.w};
#pragma unroll
    for (int h = 0; h < H_HEADS; ++h) {
        float x = v[h];
        x = x > 0.f ? x : NEG_SLOPE * x;
        atomicMax(&menc[d * 4 + h], enc_ord(x));
    }
}

// ============================================================
// K5 (edge pass B): one wave per edge.
// ex = exp(e - m); atomicAdd denom; atomicAdd out += ex * feat_src[src]
// (unnormalized accumulation — normalize in K6)
// ============================================================
__global__ __launch_bounds__(256) void gat_edge_agg(const float* __restrict__ feat_src,
                                                    const float* __restrict__ el,
                                                    const float* __restrict__ er,
                                                    const int* __restrict__ src,
                                                    const int* __restrict__ dst,
                                                    const unsigned* __restrict__ menc,
                                                    float* __restrict__ denom,
                                                    float* __restrict__ out,
                                                    int E) {
    int wid = (blockIdx.x * blockDim.x + threadIdx.x) >> 5;
    int lane = threadIdx.x & 31;
    if (wid >= E) return;
    int s = src[wid], d = dst[wid];

    const float4 l4 = *(const float4*)(el + 4 * (size_t)s);   // broadcast loads
    const float4 r4 = *(const float4*)(er + 4 * (size_t)d);
    float v[4] = {l4.x + r4.x, l4.y + r4.y, l4.z + r4.z, l4.w + r4.w};
    float ex[4];
#pragma unroll
    for (int h = 0; h < H_HEADS; ++h) {
        float x = v[h];
        x = x > 0.f ? x : NEG_SLOPE * x;
        float m = dec_ord(menc[d * 4 + h]);
        ex[h] = __expf(x - m);
    }
    if (lane < H_HEADS) atomicAdd(&denom[d * 4 + lane], ex[lane]);

    const float* fs = feat_src + (size_t)s * C_DIM;
    float* op = out + (size_t)d * C_DIM;
#pragma unroll
    for (int h = 0; h < H_HEADS; ++h) {
        float val = ex[h] * fs[h * F_DIM + lane];   // coalesced 128B per head
        atomicAdd(&op[h * F_DIM + lane], val);
    }
}

// ============================================================
// K6: out = out / max(denom,·) + bias
// ============================================================
__global__ __launch_bounds__(256) void gat_norm(float* __restrict__ out,
                                                const float* __restrict__ denom,
                                                const float* __restrict__ bias,
                                                int num_dst) {
    int i = blockIdx.x * blockDim.x + threadIdx.x;
    if (i >= num_dst * C_DIM) return;
    int n = i >> 7;
    int c = i & (C_DIM - 1);
    int h = c >> 5;
    float dn = denom[n * 4 + h];
    dn = dn > 0.f ? dn : 1.0f;
    out[i] = out[i] / dn + bias[c];
}

extern "C" void kernel_launch(void* const* d_in, const int* in_sizes, int n_in,
                              void* d_out, int out_size, void* d_ws, size_t ws_size,
                              hipStream_t stream) {
    const float* feat   = (const float*)d_in[0];
    const float* Wfc    = (const float*)d_in[1];
    const float* attn_l = (const float*)d_in[2];
    const float* attn_r = (const float*)d_in[3];
    const float* bias   = (const float*)d_in[4];
    const int*   src    = (const int*)d_in[5];
    const int*   dst    = (const int*)d_in[6];

    int N       = in_sizes[0] / C_DIM;
    int E       = in_sizes[5];
    int num_dst = out_size / C_DIM;
    float* out  = (float*)d_out;

    char* ws = (char*)d_ws;
    float*    feat_src = (float*)ws;    ws += (size_t)N * C_DIM * sizeof(float);
    float*    el       = (float*)ws;    ws += (size_t)N * H_HEADS * sizeof(float);
    float*    er       = (float*)ws;    ws += (size_t)num_dst * H_HEADS * sizeof(float);
    unsigned* menc     = (unsigned*)ws; ws += (size_t)num_dst * H_HEADS * sizeof(unsigned);
    float*    denom    = (float*)ws;

    // K1: projection (WMMA f32): 8 waves/block -> 128 rows/block
    gat_gemm<<<(N + 127) / 128, 256, 0, stream>>>(feat, Wfc, feat_src, N);
    // K2: attention logits
    gat_attn<<<(N * H_HEADS + 255) / 256, 256, 0, stream>>>(feat_src, attn_l, attn_r,
                                                            el, er, N, num_dst);
    // K3: init accumulators
    gat_init<<<(num_dst * C_DIM + 255) / 256, 256, 0, stream>>>(out, menc, denom, num_dst);
    // K4: segment max over edges
    gat_edge_max<<<(E + 255) / 256, 256, 0, stream>>>(el, er, src, dst, menc, E);
    // K5: fused exp + denom + weighted scatter (1 wave / edge, 8 waves / block)
    gat_edge_agg<<<(E + 7) / 8, 256, 0, stream>>>(feat_src, el, er, src, dst,
                                                  menc, denom, out, E);
    // K6: normalize + bias
    gat_norm<<<(num_dst * C_DIM + 255) / 256, 256, 0, stream>>>(out, denom, bias, num_dst);
}